// YOLOv2_84361747628427
// MI455X (gfx1250) — compile-verified
//
#include <hip/hip_runtime.h>
#include <math.h>

// Problem constants (fixed by reference setup_inputs)
static constexpr int Bn  = 32;     // batch
static constexpr int Hh  = 19;
static constexpr int Ww  = 19;
static constexpr int Aa  = 5;
static constexpr int Cc  = 80;
static constexpr int Mm  = Hh * Ww * Aa;      // 1805 detections per batch
static constexpr int NP  = 2048;              // padded sort size (pow2 >= Mm)
static constexpr int HW  = Hh * Ww;           // 361
static constexpr int BM  = Bn * Mm;           // 57760

// d_out layout (float32, concatenated in reference return order):
//   [0, 4*BM)        boxes_s   (B, M, 4)
//   [4*BM, 5*BM)     scores_s  (B, M)
//   [5*BM, 6*BM)     labels_s  (B, M)  (argmax written as float)
//   [6*BM, 7*BM)     keep      (B, M)  (0.0/1.0; temporarily holds valid_s)

__global__ __launch_bounds__(256) void yolo_decode_sort(
    const float* __restrict__ x, const float* __restrict__ anchors,
    float* __restrict__ out)
{
    __shared__ float skey[NP];
    __shared__ int   sidx[NP];
    __shared__ float dX1[Mm], dY1[Mm], dX2[Mm], dY2[Mm];
    __shared__ float dScore[Mm];
    __shared__ int   dMeta[Mm];   // (label << 1) | valid

    const int bb  = blockIdx.x;
    const int tid = threadIdx.x;
    const float* xb = x + (size_t)bb * (Aa * (5 + Cc)) * HW;

    // ---- decode: one thread per spatial cell s, loop anchors/channels.
    //      For fixed channel, lanes read consecutive addresses (coalesced). ----
    for (int s = tid; s < HW; s += 256) {
        const int y  = s / Ww;
        const int xx = s - y * Ww;
        const float cx = (float)xx / 19.0f;
        const float cy = (float)y  / 19.0f;

        for (int a = 0; a < Aa; ++a) {
            const float* p = xb + (size_t)(a * (5 + Cc)) * HW + s;

            const float tx = p[0 * HW];
            const float ty = p[1 * HW];
            const float tw = p[2 * HW];
            const float th = p[3 * HW];
            const float to = p[4 * HW];

            const float sx = 1.0f / (1.0f + expf(-tx));
            const float sy = 1.0f / (1.0f + expf(-ty));
            const float bx = sx / 19.0f + cx;
            const float by = sy / 19.0f + cy;
            const float bw = (anchors[a * 2 + 0] / 19.0f) * expf(tw);
            const float bh = (anchors[a * 2 + 1] / 19.0f) * expf(th);
            const float bo = 1.0f / (1.0f + expf(-to));
            const int valid = (bo >= 0.5f) ? 1 : 0;

            // class max / argmax (first occurrence of max, like jnp.argmax)
            float best = p[5 * HW];
            int   lab  = 0;
            for (int c = 1; c < Cc; ++c) {
                const float v = p[(5 + c) * HW];
                if (v > best) { best = v; lab = c; }
            }

            const int m = s * Aa + a;     // matches reshape(b, h*w*A) order
            const float hw2 = bw / 2.0f, hh2 = bh / 2.0f;
            dX1[m] = fminf(fmaxf(bx - hw2, 0.0f), 1.0f);
            dY1[m] = fminf(fmaxf(by - hh2, 0.0f), 1.0f);
            dX2[m] = fminf(fmaxf(bx + hw2, 0.0f), 1.0f);
            dY2[m] = fminf(fmaxf(by + hh2, 0.0f), 1.0f);
            dScore[m] = best;
            dMeta[m]  = (lab << 1) | valid;

            skey[m] = valid ? best : -1000000000.0f;
            sidx[m] = m;
        }
    }
    // padding for bitonic sort: keys below any real key; idx keeps comparator total
    for (int t = tid; t < NP; t += 256) {
        if (t >= Mm) { skey[t] = -3.0e38f; sidx[t] = t; }
    }
    __syncthreads();

    // ---- bitonic sort: descending by key, ascending index on ties
    //      (== stable argsort(-key)) ----
    for (int k = 2; k <= NP; k <<= 1) {
        for (int j = k >> 1; j > 0; j >>= 1) {
            for (int i = tid; i < NP; i += 256) {
                const int ixj = i ^ j;
                if (ixj > i) {
                    const float ka = skey[i], kb = skey[ixj];
                    const int   ia = sidx[i], ib = sidx[ixj];
                    const bool before = (ka > kb) || (ka == kb && ia < ib);
                    const bool want   = ((i & k) == 0);
                    if (before != want) {
                        skey[i] = kb; skey[ixj] = ka;
                        sidx[i] = ib; sidx[ixj] = ia;
                    }
                }
            }
            __syncthreads();
        }
    }

    // ---- gather in sorted order and write outputs ----
    for (int t = tid; t < Mm; t += 256) {
        const int o = sidx[t];
        const int g = bb * Mm + t;
        out[(size_t)g * 4 + 0] = dX1[o];
        out[(size_t)g * 4 + 1] = dY1[o];
        out[(size_t)g * 4 + 2] = dX2[o];
        out[(size_t)g * 4 + 3] = dY2[o];
        out[4 * BM + g] = dScore[o];
        const int meta  = dMeta[o];
        out[5 * BM + g] = (float)(meta >> 1);
        out[6 * BM + g] = (float)(meta & 1);   // valid_s (temporary)
    }
}

__global__ __launch_bounds__(256) void yolo_nms(float* __restrict__ out)
{
    __shared__ float4 lbox[Mm];
    __shared__ float  larea[Mm];
    __shared__ int    lkeep[Mm];

    const int bb  = blockIdx.x;
    const int tid = threadIdx.x;
    const float4* gbox = (const float4*)out + (size_t)bb * Mm;

    // ---- gfx1250 async global->LDS staging of this batch's sorted boxes ----
    for (int t = tid; t < Mm; t += 256) {
        const unsigned ldsa = (unsigned)(unsigned long long)(const void*)&lbox[t];
        const unsigned long long ga = (unsigned long long)(const void*)&gbox[t];
        asm volatile("global_load_async_to_lds_b128 %0, %1, off"
                     :: "v"(ldsa), "v"(ga) : "memory");
    }
    asm volatile("s_wait_asynccnt 0" ::: "memory");
    __syncthreads();

    for (int t = tid; t < Mm; t += 256) {
        const float4 b = lbox[t];
        larea[t] = (b.z - b.x) * (b.w - b.y);
        lkeep[t] = 1;
    }
    __syncthreads();

    // ---- exact greedy suppression matching the reference fori_loop ----
    for (int i = 0; i < Mm; ++i) {
        if (lkeep[i]) {
            const float4 bi = lbox[i];
            const float  ai = larea[i];
            for (int j = i + 1 + tid; j < Mm; j += 256) {
                if (!lkeep[j]) continue;
                const float4 bj = lbox[j];
                const float ltx = fmaxf(bi.x, bj.x);
                const float lty = fmaxf(bi.y, bj.y);
                const float rbx = fminf(bi.z, bj.z);
                const float rby = fminf(bi.w, bj.w);
                const float wv = fmaxf(rbx - ltx, 0.0f);
                const float hv = fmaxf(rby - lty, 0.0f);
                const float inter = wv * hv;
                const float iou = inter / (ai + larea[j] - inter);
                if (iou > 0.7f) lkeep[j] = 0;
            }
        }
        __syncthreads();
    }

    // ---- final keep = nms_keep & valid_s & (score >= 0.05) ----
    for (int t = tid; t < Mm; t += 256) {
        const int g = bb * Mm + t;
        const float sc = out[4 * BM + g];
        const float va = out[6 * BM + g];   // valid_s parked here by kernel 1
        const int k = (lkeep[t] != 0) && (va != 0.0f) && (sc >= 0.05f);
        out[6 * BM + g] = (float)k;
    }
}

extern "C" void kernel_launch(void* const* d_in, const int* in_sizes, int n_in,
                              void* d_out, int out_size, void* d_ws, size_t ws_size,
                              hipStream_t stream)
{
    (void)in_sizes; (void)n_in; (void)out_size; (void)d_ws; (void)ws_size;
    const float* x       = (const float*)d_in[0];
    const float* anchors = (const float*)d_in[1];
    float* out = (float*)d_out;

    yolo_decode_sort<<<Bn, 256, 0, stream>>>(x, anchors, out);
    yolo_nms<<<Bn, 256, 0, stream>>>(out);
}